// dynamicGAT_47820165873710
// MI455X (gfx1250) — compile-verified
//
#include <hip/hip_runtime.h>
#include <math.h>

#define NNODES 50000
#define NEDGES 800000
#define NTOT   (NEDGES + NNODES)   // 850000 edges incl. self loops
#define FEAT 128
#define HID  128
#define HEADS 8
#define HDIM 16
#define NEG_SLOPE 0.2f

typedef __attribute__((ext_vector_type(2))) float v2f;
typedef __attribute__((ext_vector_type(8))) float v8f;

static __device__ __forceinline__ void atomicMaxF(float* addr, float val) {
    if (val >= 0.f) atomicMax((int*)addr, __float_as_int(val));
    else            atomicMin((unsigned int*)addr, (unsigned int)__float_as_int(val));
}

static __device__ __forceinline__ void edge_sd(const int* __restrict__ ei, int e,
                                               int& s, int& d) {
    if (e < NEDGES) { s = ei[e]; d = ei[NEDGES + e]; }
    else            { s = e - NEDGES; d = s; }
}

// ---------------------------------------------------------------------------
// GEMM: Hout[N,128] = X[N,128] @ W[128,128] via V_WMMA_F32_16X16X4_F32.
// One 16x16 output tile per wave; 8 waves/block cover all 128 columns;
// 3125 blocks cover all 50000 rows exactly. W staged in LDS (64 KB).
// ---------------------------------------------------------------------------
__global__ __launch_bounds__(256)
void k_gemm_wmma(const float* __restrict__ X, const float* __restrict__ W,
                 float* __restrict__ Hout) {
    __shared__ float ldsW[FEAT * HID];          // 64 KB
    const int tid = threadIdx.x;

    // cooperative load of W into LDS (4096 float4 / 256 threads = 16 each)
    const float4* Wv = (const float4*)W;
    float4*       Lv = (float4*)ldsW;
#pragma unroll
    for (int i = 0; i < (FEAT * HID / 4) / 256; ++i)
        Lv[tid + i * 256] = Wv[tid + i * 256];
    __syncthreads();

    const int lane = tid & 31;
    const int wave = tid >> 5;
    const int l16  = lane & 15;
    const int half = lane >> 4;                 // 0: lanes 0-15, 1: lanes 16-31
    const int row  = blockIdx.x * 16 + l16;     // A-matrix row for this lane
    const int col  = wave * 16 + l16;           // B/C/D column for this lane

    const float* xrow = X + (size_t)row * FEAT;

    v8f acc = {};
#pragma unroll
    for (int k0 = 0; k0 < FEAT; k0 += 4) {
        const int ka = k0 + 2 * half;           // K layout per ISA 7.12.2
        v2f a, b;
        a.x = xrow[ka];
        a.y = xrow[ka + 1];
        b.x = ldsW[ka * HID + col];
        b.y = ldsW[(ka + 1) * HID + col];
        // (neg_a, A, neg_b, B, c_mod, C, reuse_a, reuse_b)
        acc = __builtin_amdgcn_wmma_f32_16x16x4_f32(
                  false, a, false, b, (short)0, acc, false, false);
    }

    const int orow0 = blockIdx.x * 16 + 8 * half;   // C/D row layout
#pragma unroll
    for (int j = 0; j < 8; ++j)
        Hout[(size_t)(orow0 + j) * HID + col] = acc[j];
}

// ---------------------------------------------------------------------------
// Per-(node,head) attention scalars: AS = <h, a_src>, AD = <h, a_dst>
// ---------------------------------------------------------------------------
__global__ __launch_bounds__(256)
void k_alpha(const float* __restrict__ Hf, const float* __restrict__ a_src,
             const float* __restrict__ a_dst, float* __restrict__ AS,
             float* __restrict__ AD) {
    int t = blockIdx.x * blockDim.x + threadIdx.x;     // n*HEADS + head
    if (t >= NNODES * HEADS) return;
    int hd = t & (HEADS - 1);
    int n  = t >> 3;
    const float* hp = Hf + (size_t)n * HID + hd * HDIM;
    const float* as = a_src + hd * HDIM;
    const float* ad = a_dst + hd * HDIM;
    float sa = 0.f, sd = 0.f;
#pragma unroll
    for (int c = 0; c < HDIM; ++c) {
        float v = hp[c];
        sa += v * as[c];
        sd += v * ad[c];
    }
    AS[t] = sa;
    AD[t] = sd;
}

__global__ __launch_bounds__(256)
void k_fill(float* __restrict__ p, float v, int n) {
    int i = blockIdx.x * blockDim.x + threadIdx.x;
    if (i < n) p[i] = v;
}

static __device__ __forceinline__ float edge_e(const float* AS, const float* AD,
                                               int s, int d, int hd) {
    float v = AS[s * HEADS + hd] + AD[d * HEADS + hd];
    return v >= 0.f ? v : NEG_SLOPE * v;                 // leaky_relu(0.2)
}

// pass 1: segment max over dst
__global__ __launch_bounds__(256)
void k_edge_max(const int* __restrict__ ei, const float* __restrict__ AS,
                const float* __restrict__ AD, float* __restrict__ M) {
    int t = blockIdx.x * blockDim.x + threadIdx.x;
    if (t >= NTOT * HEADS) return;
    int hd = t & (HEADS - 1), e = t >> 3;
    int s, d; edge_sd(ei, e, s, d);
    atomicMaxF(&M[d * HEADS + hd], edge_e(AS, AD, s, d, hd));
}

// pass 2: segment sum of exp(e - m)
__global__ __launch_bounds__(256)
void k_edge_sum(const int* __restrict__ ei, const float* __restrict__ AS,
                const float* __restrict__ AD, const float* __restrict__ M,
                float* __restrict__ S) {
    int t = blockIdx.x * blockDim.x + threadIdx.x;
    if (t >= NTOT * HEADS) return;
    int hd = t & (HEADS - 1), e = t >> 3;
    int s, d; edge_sd(ei, e, s, d);
    float w = expf(edge_e(AS, AD, s, d, hd) - M[d * HEADS + hd]);
    atomicAdd(&S[d * HEADS + hd], w);
}

// pass 3: weighted scatter of h[src] into Agg[dst]
__global__ __launch_bounds__(256)
void k_edge_agg(const int* __restrict__ ei, const float* __restrict__ AS,
                const float* __restrict__ AD, const float* __restrict__ M,
                const float* __restrict__ S, const float* __restrict__ Hf,
                float* __restrict__ Agg) {
    int t = blockIdx.x * blockDim.x + threadIdx.x;
    if (t >= NTOT * HEADS) return;
    int hd = t & (HEADS - 1), e = t >> 3;
    int s, d; edge_sd(ei, e, s, d);
    float w     = expf(edge_e(AS, AD, s, d, hd) - M[d * HEADS + hd]);
    float alpha = w / (S[d * HEADS + hd] + 1e-16f);
    const float* hs = Hf  + (size_t)s * HID + hd * HDIM;
    float*       op = Agg + (size_t)d * HID + hd * HDIM;
    __builtin_prefetch(hs, 0, 1);                       // global_prefetch_b8
#pragma unroll
    for (int c = 0; c < HDIM; ++c)
        atomicAdd(&op[c], alpha * hs[c]);               // global_atomic_add_f32
}

// out = elu(agg + b) (in place allowed: same element per thread)
__global__ __launch_bounds__(256)
void k_bias_elu(const float* __restrict__ Agg, const float* __restrict__ b,
                float* __restrict__ Out) {
    int t = blockIdx.x * blockDim.x + threadIdx.x;
    if (t >= NNODES * HID) return;
    float v = Agg[t] + b[t & (HID - 1)];
    Out[t] = v > 0.f ? v : expm1f(v);
}

static inline unsigned grid_for(long long n) { return (unsigned)((n + 255) / 256); }

extern "C" void kernel_launch(void* const* d_in, const int* in_sizes, int n_in,
                              void* d_out, int out_size, void* d_ws, size_t ws_size,
                              hipStream_t stream) {
    const float* x   = (const float*)d_in[0];
    const int*   ei  = (const int*)  d_in[1];
    const float* W1  = (const float*)d_in[2];
    const float* as1 = (const float*)d_in[3];
    const float* ad1 = (const float*)d_in[4];
    const float* b1  = (const float*)d_in[5];
    const float* W2  = (const float*)d_in[6];
    const float* as2 = (const float*)d_in[7];
    const float* ad2 = (const float*)d_in[8];
    const float* b2  = (const float*)d_in[9];

    float* ws     = (float*)d_ws;
    float* bufH   = ws;                               // N*HID      = 6.4M
    float* bufAS  = bufH  + (size_t)NNODES * HID;     // N*HEADS
    float* bufAD  = bufAS + (size_t)NNODES * HEADS;
    float* bufM   = bufAD + (size_t)NNODES * HEADS;
    float* bufS   = bufM  + (size_t)NNODES * HEADS;
    float* bufAgg = bufS  + (size_t)NNODES * HEADS;   // N*HID (layer1 out / layer2 in)

    const float neg_inf = -__builtin_huge_valf();

    auto run_layer = [&](const float* xin, const float* Wm, const float* as,
                         const float* ad, const float* bv, float* outBuf) {
        k_gemm_wmma<<<NNODES / 16, 256, 0, stream>>>(xin, Wm, bufH);
        k_alpha<<<grid_for((long long)NNODES * HEADS), 256, 0, stream>>>(
            bufH, as, ad, bufAS, bufAD);
        k_fill<<<grid_for((long long)NNODES * HEADS), 256, 0, stream>>>(
            bufM, neg_inf, NNODES * HEADS);
        hipMemsetAsync(bufS, 0, (size_t)NNODES * HEADS * sizeof(float), stream);
        hipMemsetAsync(outBuf, 0, (size_t)NNODES * HID * sizeof(float), stream);
        k_edge_max<<<grid_for((long long)NTOT * HEADS), 256, 0, stream>>>(
            ei, bufAS, bufAD, bufM);
        k_edge_sum<<<grid_for((long long)NTOT * HEADS), 256, 0, stream>>>(
            ei, bufAS, bufAD, bufM, bufS);
        k_edge_agg<<<grid_for((long long)NTOT * HEADS), 256, 0, stream>>>(
            ei, bufAS, bufAD, bufM, bufS, bufH, outBuf);
        k_bias_elu<<<grid_for((long long)NNODES * HID), 256, 0, stream>>>(
            outBuf, bv, outBuf);
    };

    run_layer(x,      W1, as1, ad1, b1, bufAgg);        // layer 1 -> bufAgg
    run_layer(bufAgg, W2, as2, ad2, b2, (float*)d_out); // layer 2 -> d_out
}